// DIGAT_12541304504427
// MI455X (gfx1250) — compile-verified
//
#include <hip/hip_runtime.h>
#include <math.h>

#define D_    400
#define B_    64
#define N_    31
#define H_    50
#define CAT_  17
#define C_    18
#define U_    67
#define DEPTH_ 3
#define NEGV  (-1000000000.0f)

typedef float v2f __attribute__((ext_vector_type(2)));
typedef float v8f __attribute__((ext_vector_type(8)));

// ---------------------------------------------------------------------------
// C[M,Nn] = A @ W^T (+bias).  A row m lives at:
//   A + (m / row_inner) * row_outer + (m % row_inner) * lda
// (row_inner = M, row_outer = 0 for dense A). W is (Nn, K) row-major.
//
// One wave computes a 32x80 output tile = 2 (M) x 5 (N) WMMA subtiles:
// per K-step, 2 A-fragments + 5 B-fragments feed 10 V_WMMA_F32_16X16X4_F32.
// The K-loop is software-pipelined: the next step's 7 fragment loads are
// issued into fresh registers before the current step's 10 WMMAs, so global
// latency overlaps the matrix pipe. __launch_bounds__(128, 1) relaxes the
// VGPR cap (wave32 allows up to 1024) so the double-buffered fragments and
// 80 accumulator VGPRs all stay in registers.
//
// Requirements (all satisfied by this model): M % 32 == 0, Nn % 80 == 0,
// K % 4 == 0. EXEC all-ones (branching is wave-uniform).
//
// f32 16x16x4 fragment layout (ISA 7.12.2):
//   A (16x4):  lanes 0-15: M=lane, v0=K0 v1=K1 ; lanes 16-31: M=lane-16, v0=K2 v1=K3
//   B (4x16):  lanes 0-15: N=lane, v0=K0 v1=K1 ; lanes 16-31: N=lane-16, v0=K2 v1=K3
//   C (16x16): vgpr v: lanes 0-15 -> M=v, lanes 16-31 -> M=v+8; N=lane%16
// ---------------------------------------------------------------------------
__global__ __launch_bounds__(128, 1)
void gemm_wmma(const float* __restrict__ A, int lda, int row_inner, long row_outer,
               const float* __restrict__ W, int ldw,
               const float* __restrict__ bias,
               float* __restrict__ C, int ldc,
               int M, int Nn, int K)
{
  const int tilesN = Nn / 80;
  const int totalTiles = (M >> 5) * tilesN;
  const int wavesPerBlock = blockDim.x >> 5;
  int tile = blockIdx.x * wavesPerBlock + ((int)threadIdx.x >> 5);
  if (tile >= totalTiles) return;                 // wave-uniform exit
  const int tileM = tile / tilesN;
  const int tileN = tile % tilesN;
  const int lane = threadIdx.x & 31;
  const int half = lane >> 4;                     // 0: K0/K1, 1: K2/K3
  const int r    = lane & 15;

  const int mrow0 = tileM * 32 + r;
  const int mrow1 = mrow0 + 16;
  const int koff = 2 * half;
  const float* Ap0 = A + (long)(mrow0 / row_inner) * row_outer
                       + (long)(mrow0 % row_inner) * lda + koff;
  const float* Ap1 = A + (long)(mrow1 / row_inner) * row_outer
                       + (long)(mrow1 % row_inner) * lda + koff;
  const float* Wp[5];
#pragma unroll
  for (int j = 0; j < 5; ++j)
    Wp[j] = W + (long)(tileN * 80 + j * 16 + r) * ldw + koff;

  v8f acc[2][5];
#pragma unroll
  for (int j = 0; j < 5; ++j) {
    float binit = bias ? bias[tileN * 80 + j * 16 + r] : 0.0f;
#pragma unroll
    for (int v = 0; v < 8; ++v) { acc[0][j][v] = binit; acc[1][j][v] = binit; }
  }

  // software pipeline: cur fragments in registers, next loads in flight
  v2f a0, a1, bb[5];
  a0 = *(const v2f*)(Ap0);
  a1 = *(const v2f*)(Ap1);
#pragma unroll
  for (int j = 0; j < 5; ++j) bb[j] = *(const v2f*)(Wp[j]);

  for (int k = 0; k < K - 4; k += 4) {
    v2f na0, na1, nbb[5];
    na0 = *(const v2f*)(Ap0 + k + 4);
    na1 = *(const v2f*)(Ap1 + k + 4);
#pragma unroll
    for (int j = 0; j < 5; ++j) nbb[j] = *(const v2f*)(Wp[j] + k + 4);

#pragma unroll
    for (int j = 0; j < 5; ++j) {
      acc[0][j] = __builtin_amdgcn_wmma_f32_16x16x4_f32(false, a0, false, bb[j],
                                                        (short)0, acc[0][j], false, false);
      acc[1][j] = __builtin_amdgcn_wmma_f32_16x16x4_f32(false, a1, false, bb[j],
                                                        (short)0, acc[1][j], false, false);
    }
    a0 = na0; a1 = na1;
#pragma unroll
    for (int j = 0; j < 5; ++j) bb[j] = nbb[j];
  }
#pragma unroll
  for (int j = 0; j < 5; ++j) {
    acc[0][j] = __builtin_amdgcn_wmma_f32_16x16x4_f32(false, a0, false, bb[j],
                                                      (short)0, acc[0][j], false, false);
    acc[1][j] = __builtin_amdgcn_wmma_f32_16x16x4_f32(false, a1, false, bb[j],
                                                      (short)0, acc[1][j], false, false);
  }

#pragma unroll
  for (int mi = 0; mi < 2; ++mi) {
#pragma unroll
    for (int j = 0; j < 5; ++j) {
#pragma unroll
      for (int v = 0; v < 8; ++v) {
        int m = tileM * 32 + mi * 16 + v + 8 * half;
        C[(long)m * ldc + tileN * 80 + j * 16 + r] = acc[mi][j][v];
      }
    }
  }
}

// ---------------------------------------------------------------------------
// SDP attention: a[n] = <K[b,n,:], Q[b,:]> / 20; mask==0 -> NEG; softmax over n;
// out[b,:] = sum_n alpha[n] * feat[b,n,:].   n <= 32. One block per batch.
// ---------------------------------------------------------------------------
__global__ void sdp_combine(const float* __restrict__ K, const float* __restrict__ Q,
                            const float* __restrict__ feat, const int* __restrict__ mask,
                            float* __restrict__ out, int n)
{
  const int b = blockIdx.x;
  __shared__ float alpha[32];
  const int tid = threadIdx.x;
  if (tid < 32) {
    float s = NEGV;
    if (tid < n && mask[b * n + tid] != 0) {
      const float* kp = K + ((long)b * n + tid) * D_;
      const float* qp = Q + (long)b * D_;
      float acc = 0.f;
      for (int d = 0; d < D_; ++d) acc += kp[d] * qp[d];
      s = acc * 0.05f;                            // 1/sqrt(400)
    }
    float m = s;
    for (int off = 16; off; off >>= 1) m = fmaxf(m, __shfl_xor(m, off, 32));
    float e = __expf(s - m);
    float sum = e;
    for (int off = 16; off; off >>= 1) sum += __shfl_xor(sum, off, 32);
    alpha[tid] = e / sum;
  }
  __syncthreads();
  for (int d = tid; d < D_; d += blockDim.x) {
    float acc = 0.f;
    for (int j = 0; j < n; ++j) acc += alpha[j] * feat[((long)b * n + j) * D_ + d];
    out[(long)b * D_ + d] = acc;
  }
}

// uge = concat(user_news_embedding (B,H,D), broadcast tne (CAT,D)) -> (B,U,D)
__global__ void build_uge(const float* __restrict__ une, const float* __restrict__ tne,
                          float* __restrict__ uge)
{
  long i = (long)blockIdx.x * blockDim.x + threadIdx.x;
  const long total = (long)B_ * U_ * D_;
  if (i >= total) return;
  int d = (int)(i % D_);
  long bu = i / D_;
  int u = (int)(bu % U_);
  int b = (int)(bu / U_);
  uge[i] = (u < H_) ? une[((long)b * H_ + u) * D_ + d]
                    : tne[(long)(u - H_) * D_ + d];
}

// cc[b, :] = [ nge[b,0,:], glob[b,:] ]  (B, 2D)
__global__ void concat_local_glob(const float* __restrict__ nge, const float* __restrict__ glob,
                                  float* __restrict__ cc)
{
  int i = blockIdx.x * blockDim.x + threadIdx.x;
  if (i >= B_ * 2 * D_) return;
  int b = i / (2 * D_);
  int d = i % (2 * D_);
  cc[i] = (d < D_) ? nge[((long)b * N_) * D_ + d]
                   : glob[(long)b * D_ + (d - D_)];
}

// ctx = sigmoid(gpre) * local + (1 - sigmoid(gpre)) * glob
__global__ void gate_combine(const float* __restrict__ gpre, const float* __restrict__ nge,
                             const float* __restrict__ glob, float* __restrict__ ctx)
{
  int i = blockIdx.x * blockDim.x + threadIdx.x;
  if (i >= B_ * D_) return;
  int b = i / D_;
  int d = i % D_;
  float g = 1.f / (1.f + __expf(-gpre[i]));
  float local = nge[((long)b * N_) * D_ + d];
  ctx[i] = g * local + (1.f - g) * glob[i];
}

// ---------------------------------------------------------------------------
// Segmented (per-category) softmax over history + topic aggregation.
// One block per batch. hist = uge[:, :H, :].
// ---------------------------------------------------------------------------
__global__ void user_seg_topic(const float* __restrict__ K, const float* __restrict__ Q,
                               const float* __restrict__ uge, const int* __restrict__ cidx,
                               float* __restrict__ topic)
{
  const int b = blockIdx.x;
  __shared__ float a_sh[H_];
  __shared__ float e_sh[H_];
  __shared__ int   i_sh[H_];
  __shared__ float segmax[C_];
  __shared__ float segsum[C_];
  const int tid = threadIdx.x;

  if (tid < H_) {
    const float* kp = K + ((long)b * H_ + tid) * D_;
    const float* qp = Q + (long)b * D_;
    float acc = 0.f;
    for (int d = 0; d < D_; ++d) acc += kp[d] * qp[d];
    a_sh[tid] = acc * 0.05f;
    i_sh[tid] = cidx[b * H_ + tid];
  }
  __syncthreads();
  if (tid < C_) {
    float m = -INFINITY;
    for (int h = 0; h < H_; ++h) if (i_sh[h] == tid) m = fmaxf(m, a_sh[h]);
    segmax[tid] = m;
  }
  __syncthreads();
  if (tid < H_) e_sh[tid] = __expf(a_sh[tid] - segmax[i_sh[tid]]);
  __syncthreads();
  if (tid < C_) {
    float s = 0.f;
    for (int h = 0; h < H_; ++h) if (i_sh[h] == tid) s += e_sh[h];
    segsum[tid] = s;
  }
  __syncthreads();
  if (tid < H_) e_sh[tid] = e_sh[tid] / segsum[i_sh[tid]];
  __syncthreads();

  for (int d = tid; d < D_; d += blockDim.x) {
    float acc[C_];
#pragma unroll
    for (int c = 0; c < C_; ++c) acc[c] = 0.f;
    for (int h = 0; h < H_; ++h) {
      float v = e_sh[h] * uge[((long)b * U_ + h) * D_ + d];
      int ci = i_sh[h];
#pragma unroll
      for (int c = 0; c < C_; ++c) acc[c] += (ci == c) ? v : 0.f;
    }
#pragma unroll
    for (int c = 0; c < C_; ++c) topic[((long)b * C_ + c) * D_ + d] = acc[c];
  }
}

// y = relu(y) + res (elementwise)
__global__ void relu_residual(float* __restrict__ y, const float* __restrict__ res, long n)
{
  long i = (long)blockIdx.x * blockDim.x + threadIdx.x;
  if (i < n) y[i] = fmaxf(y[i], 0.f) + res[i];
}

// ---------------------------------------------------------------------------
// GAT attention: one block per (b,i).
// s[j] = sum_d relu(xf1[b,j,d] + xf2[b,i,d] + K3[b,d]) * aw[d]
// e = leaky(s, 0.2); graph==0 -> NEG; softmax over j;
// out[b,i,:] = relu(sum_j alpha[j] * h[b,j,:]) + x[b,i,:]
// ---------------------------------------------------------------------------
__global__ void gat_attn(const float* __restrict__ xf1, const float* __restrict__ xf2,
                         const float* __restrict__ K3, const float* __restrict__ aw,
                         const float* __restrict__ h, const float* __restrict__ x,
                         const int* __restrict__ graph, float* __restrict__ out, int n)
{
  const int bi = blockIdx.x;          // b*n + i
  const int b = bi / n;
  const int i = bi % n;
  __shared__ float pre[D_];
  __shared__ float awsh[D_];
  __shared__ float s_sh[80];          // n <= 67
  const int tid = threadIdx.x;

  for (int d = tid; d < D_; d += blockDim.x) {
    pre[d]  = xf2[(long)bi * D_ + d] + K3[(long)b * D_ + d];
    awsh[d] = aw[d];
  }
  __syncthreads();

  const int wave = tid >> 5, lane = tid & 31;
  const int nw = blockDim.x >> 5;
  for (int j = wave; j < n; j += nw) {
    const float* f1p = xf1 + ((long)b * n + j) * D_;
    float p = 0.f;
    for (int d = lane; d < D_; d += 32) {
      float t = f1p[d] + pre[d];
      p += fmaxf(t, 0.f) * awsh[d];
    }
    for (int off = 16; off; off >>= 1) p += __shfl_xor(p, off, 32);
    if (lane == 0) {
      float e = (p >= 0.f) ? p : 0.2f * p;
      if (graph[((long)b * n + i) * n + j] == 0) e = NEGV;
      s_sh[j] = e;
    }
  }
  __syncthreads();

  if (tid < 32) {
    float m = -INFINITY;
    for (int j = tid; j < n; j += 32) m = fmaxf(m, s_sh[j]);
    for (int off = 16; off; off >>= 1) m = fmaxf(m, __shfl_xor(m, off, 32));
    float sum = 0.f;
    for (int j = tid; j < n; j += 32) sum += __expf(s_sh[j] - m);
    for (int off = 16; off; off >>= 1) sum += __shfl_xor(sum, off, 32);
    for (int j = tid; j < n; j += 32) s_sh[j] = __expf(s_sh[j] - m) / sum;
  }
  __syncthreads();

  for (int d = tid; d < D_; d += blockDim.x) {
    float acc = 0.f;
    for (int j = 0; j < n; ++j) acc += s_sh[j] * h[((long)b * n + j) * D_ + d];
    out[(long)bi * D_ + d] = fmaxf(acc, 0.f) + x[(long)bi * D_ + d];
  }
}

// out[b, :] = [ ng_ctx[b,:], ug_ctx[b,:] ]
__global__ void final_concat(const float* __restrict__ ngc, const float* __restrict__ ugc,
                             float* __restrict__ out)
{
  int i = blockIdx.x * blockDim.x + threadIdx.x;
  if (i >= B_ * 2 * D_) return;
  int b = i / (2 * D_);
  int d = i % (2 * D_);
  out[i] = (d < D_) ? ngc[(long)b * D_ + d] : ugc[(long)b * D_ + d - D_];
}

// ---------------------------------------------------------------------------
extern "C" void kernel_launch(void* const* d_in, const int* in_sizes, int n_in,
                              void* d_out, int out_size, void* d_ws, size_t ws_size,
                              hipStream_t stream)
{
  (void)in_sizes; (void)n_in; (void)out_size; (void)ws_size;

  const float* in_nge    = (const float*)d_in[0];
  const int*   news_graph= (const int*)d_in[1];
  const int*   ng_mask   = (const int*)d_in[2];
  const float* une       = (const float*)d_in[3];
  const int*   user_graph= (const int*)d_in[4];
  const int*   cat_mask  = (const int*)d_in[5];
  const int*   cat_idx   = (const int*)d_in[6];

  const float* P[28];
  for (int k = 0; k < 28; ++k) P[k] = (const float*)d_in[7 + k];
  const float* tne     = P[0];
  const float* cand_K  = P[1];
  const float* cand_Qw = P[2];
  const float* cand_Qb = P[3];
  const float* ngW_w   = P[4];
  const float* ngW_b   = P[5];
  const float* unK     = P[6];
  const float* unQw    = P[7];
  const float* unQb    = P[8];
  const float* aff_w   = P[9];
  const float* aff_b   = P[10];
  const float* uatt_K  = P[11];
  const float* uatt_Qw = P[12];
  const float* uatt_Qb = P[13];
  const float* n_W_w   = P[14];
  const float* n_W_b   = P[15];
  const float* n_f1    = P[16];
  const float* n_f2    = P[17];
  const float* n_f3w   = P[18];
  const float* n_f3b   = P[19];
  const float* n_a     = P[20];
  const float* u_W_w   = P[21];
  const float* u_W_b   = P[22];
  const float* u_f1    = P[23];
  const float* u_f2    = P[24];
  const float* u_f3w   = P[25];
  const float* u_f3b   = P[26];
  const float* u_a     = P[27];

  float* ws = (float*)d_ws;
  size_t off = 0;
  auto alloc = [&](size_t nElems) { float* p = ws + off; off += nElems; return p; };
  float* ugeA  = alloc((size_t)B_ * U_ * D_);
  float* ugeB  = alloc((size_t)B_ * U_ * D_);
  float* ngeA  = alloc((size_t)B_ * N_ * D_);
  float* ngeB  = alloc((size_t)B_ * N_ * D_);
  float* buf1  = alloc((size_t)B_ * U_ * D_);   // h
  float* buf2  = alloc((size_t)B_ * U_ * D_);   // xf1 / K scratch
  float* buf3  = alloc((size_t)B_ * U_ * D_);   // xf2
  float* k3    = alloc((size_t)B_ * D_);
  float* qbuf  = alloc((size_t)B_ * D_);
  float* ngc   = alloc((size_t)B_ * D_);
  float* ugc   = alloc((size_t)B_ * D_);
  float* globb = alloc((size_t)B_ * D_);
  float* cc2   = alloc((size_t)B_ * 2 * D_);
  float* gpre  = alloc((size_t)B_ * D_);
  float* topic = alloc((size_t)B_ * C_ * D_);
  float* topicK= alloc((size_t)B_ * C_ * D_);

  auto gemm = [&](const float* A, int lda, int row_inner, long row_outer,
                  const float* W, int ldw, const float* bias,
                  float* C, int ldc, int M, int Nn, int K) {
    int tiles = (M / 32) * (Nn / 80);
    int blocks = (tiles + 3) / 4;                 // 4 waves (128 threads) / block
    gemm_wmma<<<dim3(blocks), dim3(128), 0, stream>>>(A, lda, row_inner, row_outer,
                                                      W, ldw, bias, C, ldc, M, Nn, K);
  };

  // ---- news_graph_context(ngeX) -> ngc ----
  auto news_ctx = [&](const float* ngeX) {
    gemm(ngeX, D_, B_ * N_, 0, cand_K, D_, nullptr, buf2, D_, B_ * N_, D_, D_);
    gemm(ngeX, D_, 1, (long)N_ * D_, cand_Qw, D_, cand_Qb, qbuf, D_, B_, D_, D_);
    sdp_combine<<<B_, 256, 0, stream>>>(buf2, qbuf, ngeX, ng_mask, globb, N_);
    concat_local_glob<<<(B_ * 2 * D_ + 255) / 256, 256, 0, stream>>>(ngeX, globb, cc2);
    gemm(cc2, 2 * D_, B_, 0, ngW_w, 2 * D_, ngW_b, gpre, D_, B_, D_, 2 * D_);
    gate_combine<<<(B_ * D_ + 255) / 256, 256, 0, stream>>>(gpre, ngeX, globb, ngc);
  };

  // ---- user_graph_context(ugeX, ngc) -> ugc ----
  auto user_ctx = [&](const float* ugeX) {
    gemm(ugeX, D_, H_, (long)U_ * D_, unK, D_, nullptr, buf2, D_, B_ * H_, D_, D_);
    gemm(ngc, D_, B_, 0, unQw, D_, unQb, qbuf, D_, B_, D_, D_);
    user_seg_topic<<<B_, 256, 0, stream>>>(buf2, qbuf, ugeX, cat_idx, topic);
    gemm(topic, D_, B_ * C_, 0, aff_w, D_, aff_b, topicK, D_, B_ * C_, D_, D_);
    long nt = (long)B_ * C_ * D_;
    relu_residual<<<(int)((nt + 255) / 256), 256, 0, stream>>>(topicK, topic, nt);
    gemm(topicK, D_, B_ * C_, 0, uatt_K, D_, nullptr, buf2, D_, B_ * C_, D_, D_);
    gemm(ngc, D_, B_, 0, uatt_Qw, D_, uatt_Qb, qbuf, D_, B_, D_, D_);
    sdp_combine<<<B_, 256, 0, stream>>>(buf2, qbuf, topicK, cat_mask, ugc, C_);
  };

  // ---- gat layer ----
  auto gat = [&](const float* x, const int* graph, const float* ctx,
                 const float* Ww, const float* Wb, const float* f1, const float* f2,
                 const float* f3w, const float* f3b, const float* aw,
                 int n, float* xout) {
    gemm(x, D_, B_ * n, 0, Ww, D_, Wb, buf1, D_, B_ * n, D_, D_);
    gemm(x, D_, B_ * n, 0, f1, D_, nullptr, buf2, D_, B_ * n, D_, D_);
    gemm(x, D_, B_ * n, 0, f2, D_, nullptr, buf3, D_, B_ * n, D_, D_);
    gemm(ctx, D_, B_, 0, f3w, D_, f3b, k3, D_, B_, D_, D_);
    gat_attn<<<B_ * n, 256, 0, stream>>>(buf2, buf3, k3, aw, buf1, x, graph, xout, n);
  };

  // ==== forward ====
  long tot_uge = (long)B_ * U_ * D_;
  build_uge<<<(int)((tot_uge + 255) / 256), 256, 0, stream>>>(une, tne, ugeA);

  const float* nge_cur = in_nge;
  const float* uge_cur = ugeA;

  news_ctx(nge_cur);
  user_ctx(uge_cur);

  for (int i = 0; i < DEPTH_; ++i) {
    float* nout = (i & 1) ? ngeB : ngeA;
    float* uout = (uge_cur == ugeA) ? ugeB : ugeA;
    size_t wOff = (size_t)i * D_ * D_;
    size_t bOff = (size_t)i * D_;
    gat(nge_cur, news_graph, ugc,
        n_W_w + wOff, n_W_b + bOff, n_f1 + wOff, n_f2 + wOff,
        n_f3w + wOff, n_f3b + bOff, n_a + bOff, N_, nout);
    gat(uge_cur, user_graph, ngc,
        u_W_w + wOff, u_W_b + bOff, u_f1 + wOff, u_f2 + wOff,
        u_f3w + wOff, u_f3b + bOff, u_a + bOff, U_, uout);
    nge_cur = nout;
    uge_cur = uout;
    news_ctx(nge_cur);
    user_ctx(uge_cur);
  }

  final_concat<<<(B_ * 2 * D_ + 255) / 256, 256, 0, stream>>>(ngc, ugc, (float*)d_out);
}